// VectorQuantization_47502338294577
// MI455X (gfx1250) — compile-verified
//
#include <hip/hip_runtime.h>
#include <hip/hip_bf16.h>

typedef __attribute__((ext_vector_type(2))) float v2f;
typedef __attribute__((ext_vector_type(4))) float v4f;
typedef __attribute__((ext_vector_type(8))) float v8f;

#define DDIM 256
#define KCB  1024
#define NTOK (16 * 4096)
#define ROWPAD 260           // 256 + 4 pad: distinct LDS banks across lanes, 16B-aligned rows
#define TOK_PER_BLOCK 128
#define WAVES 8
#define CTILES 4             // column tiles processed concurrently (independent WMMA chains)

// ---------------------------------------------------------------------------
// Kernel 1: c_sq[k] = ||codebook[k]||^2  (1024 values into workspace)
// ---------------------------------------------------------------------------
__global__ void vq_csq_kernel(const float* __restrict__ cb, float* __restrict__ csq) {
    int t = blockIdx.x * blockDim.x + threadIdx.x;
    if (t < KCB) {
        const v4f* r = (const v4f*)(cb + (size_t)t * DDIM);
        float s = 0.f;
#pragma unroll 8
        for (int i = 0; i < DDIM / 4; ++i) {
            v4f v = r[i];
            s += v.x * v.x + v.y * v.y + v.z * v.z + v.w * v.w;
        }
        csq[t] = s;
    }
}

// ---------------------------------------------------------------------------
// Kernel 2: main VQ — distances via V_WMMA_F32_16X16X4_F32 (4 concurrent
// accumulator chains), x tile staged via GLOBAL_LOAD_ASYNC_TO_LDS_B128,
// argmin, gather, straight-through output, loss partial.
// ---------------------------------------------------------------------------
__global__ void __launch_bounds__(256)
vq_main_kernel(const float* __restrict__ x, const float* __restrict__ cb,
               const float* __restrict__ csq, float* __restrict__ xq,
               float* __restrict__ ind_out, float* __restrict__ partial) {
    extern __shared__ float xs[];                 // TOK_PER_BLOCK rows of ROWPAD floats
    __shared__ int   widx[WAVES * 16];
    __shared__ float wsum[WAVES];

    const int tid  = threadIdx.x;
    const int wave = tid >> 5;
    const int lane = tid & 31;
    const int half = lane >> 4;                   // 0: lanes 0-15, 1: lanes 16-31
    const int idx  = lane & 15;
    const int blk_row0 = blockIdx.x * TOK_PER_BLOCK;

    // ---- Stage x tile into LDS via async global->LDS (ASYNCcnt-tracked) ----
    {
        const float* xg = x + (size_t)blk_row0 * DDIM;
        for (int r = wave; r < TOK_PER_BLOCK; r += WAVES) {
            const unsigned ldsrow = (unsigned)(uintptr_t)(xs + r * ROWPAD);
            const float*   src    = xg + (size_t)r * DDIM;
#pragma unroll
            for (int c = lane; c < DDIM / 4; c += 32) {   // 2 iterations, b128 each
                unsigned     ldsaddr = ldsrow + (unsigned)c * 16u;
                const float* gaddr   = src + c * 4;
                asm volatile("global_load_async_to_lds_b128 %0, %1, off"
                             :: "v"(ldsaddr), "v"(gaddr) : "memory");
            }
        }
        asm volatile("s_wait_asynccnt 0x0" ::: "memory");
    }
    __syncthreads();

    const int lrow0 = wave * 16;                  // this wave's 16 tokens (local)
    // A-fragment base: lane holds row (lrow0+idx), K = k0 + 2*half .. +1
    const float* arow = xs + (size_t)(lrow0 + idx) * ROWPAD + 2 * half;

    float minval[8];
    int   minidx[8];
#pragma unroll
    for (int r = 0; r < 8; ++r) { minval[r] = 3.402823e38f; minidx[r] = 0; }

    // ---- 16 groups of 4 column tiles; shared A fragment, 4 WMMA chains ----
    for (int cg = 0; cg < KCB / (16 * CTILES); ++cg) {
        const int colbase = cg * (16 * CTILES) + idx;
        const float* brow[CTILES];
        v8f acc[CTILES];
#pragma unroll
        for (int t = 0; t < CTILES; ++t) {
            brow[t] = cb + (size_t)(colbase + 16 * t) * DDIM + 2 * half;
            acc[t] = (v8f){};
        }
#pragma unroll 8
        for (int k0 = 0; k0 < DDIM; k0 += 4) {
            v2f a = *(const v2f*)(arow + k0);     // LDS, reused by CTILES WMMAs
#pragma unroll
            for (int t = 0; t < CTILES; ++t) {
                v2f b = *(const v2f*)(brow[t] + k0);
                acc[t] = __builtin_amdgcn_wmma_f32_16x16x4_f32(
                    false, a, false, b, (short)0, acc[t], false, false);
            }
        }
#pragma unroll
        for (int t = 0; t < CTILES; ++t) {
            const int col = colbase + 16 * t;
            const float cs = csq[col];
#pragma unroll
            for (int r = 0; r < 8; ++r) {
                float dist = cs - 2.0f * acc[t][r];   // ||x||^2 constant per row
                if (dist < minval[r]) { minval[r] = dist; minidx[r] = col; }
            }
        }
    }

    // ---- argmin reduction across the 16 lanes of each half (rows r+8*half) ----
#pragma unroll
    for (int r = 0; r < 8; ++r) {
        float v = minval[r];
        int   j = minidx[r];
#pragma unroll
        for (int m = 1; m <= 8; m <<= 1) {
            float ov = __shfl_xor(v, m, 32);
            int   oj = __shfl_xor(j, m, 32);
            if (ov < v || (ov == v && oj < j)) { v = ov; j = oj; }
        }
        minval[r] = v;
        minidx[r] = j;
    }
    if (idx == 0) {
#pragma unroll
        for (int r = 0; r < 8; ++r) widx[wave * 16 + 8 * half + r] = minidx[r];
    }
    __syncthreads();

    // ---- indices output (values fit exactly in float) ----
    if (lane < 16) ind_out[blk_row0 + lrow0 + lane] = (float)widx[wave * 16 + lane];

    // ---- gather codebook rows (xq == quantize), accumulate loss ----
    float lsum = 0.f;
    for (int m = 0; m < 16; ++m) {
        const int j = widx[wave * 16 + m];
        const v4f* qrow = (const v4f*)(cb + (size_t)j * DDIM);
        const v4f* xrow = (const v4f*)(xs + (size_t)(lrow0 + m) * ROWPAD);
        v4f*       orow = (v4f*)(xq + (size_t)(blk_row0 + lrow0 + m) * DDIM);
        for (int c = lane; c < DDIM / 4; c += 32) {
            v4f q  = qrow[c];
            v4f xv = xrow[c];
            __builtin_nontemporal_store(q, &orow[c]);   // write-once stream, keep L2 for cb
            v4f d = q - xv;
            lsum += d.x * d.x + d.y * d.y + d.z * d.z + d.w * d.w;
        }
    }
#pragma unroll
    for (int m = 1; m < 32; m <<= 1) lsum += __shfl_xor(lsum, m, 32);
    if (lane == 0) wsum[wave] = lsum;
    __syncthreads();
    if (tid == 0) {
        float s = 0.f;
        for (int w = 0; w < WAVES; ++w) s += wsum[w];
        partial[blockIdx.x] = s;
    }
}

// ---------------------------------------------------------------------------
// Kernel 3: deterministic reduction of per-block loss partials
// ---------------------------------------------------------------------------
__global__ void vq_loss_kernel(const float* __restrict__ partial, int nblocks,
                               float* __restrict__ loss_out) {
    __shared__ float sm[256];
    int t = threadIdx.x;
    float s = 0.f;
    for (int i = t; i < nblocks; i += 256) s += partial[i];
    sm[t] = s;
    __syncthreads();
    for (int off = 128; off > 0; off >>= 1) {
        if (t < off) sm[t] += sm[t + off];
        __syncthreads();
    }
    if (t == 0) loss_out[0] = sm[0] / 16777216.0f;   // mean over 16*4096*256
}

// ---------------------------------------------------------------------------
extern "C" void kernel_launch(void* const* d_in, const int* in_sizes, int n_in,
                              void* d_out, int out_size, void* d_ws, size_t ws_size,
                              hipStream_t stream) {
    (void)in_sizes; (void)n_in; (void)out_size; (void)ws_size;
    const float* x  = (const float*)d_in[0];
    const float* cb = (const float*)d_in[1];

    float* out  = (float*)d_out;
    float* xq   = out;                             // [NTOK * DDIM]
    float* ind  = out + (size_t)NTOK * DDIM;       // [NTOK]
    float* loss = ind + NTOK;                      // [1]

    float* csq     = (float*)d_ws;                 // [KCB]
    float* partial = csq + KCB;                    // [NTOK / TOK_PER_BLOCK]

    const int nblocks  = NTOK / TOK_PER_BLOCK;     // 512
    const size_t ldsBytes = (size_t)TOK_PER_BLOCK * ROWPAD * sizeof(float); // ~130 KB

    // Allow >64KB dynamic LDS (CDNA5 WGP has 320KB); ignore result if unsupported.
    (void)hipFuncSetAttribute((const void*)vq_main_kernel,
                              hipFuncAttributeMaxDynamicSharedMemorySize,
                              (int)ldsBytes);

    vq_csq_kernel<<<(KCB + 255) / 256, 256, 0, stream>>>(cb, csq);
    vq_main_kernel<<<nblocks, 256, ldsBytes, stream>>>(x, cb, csq, xq, ind, partial);
    vq_loss_kernel<<<1, 256, 0, stream>>>(partial, nblocks, loss);
}